// VDP_Conv2D_33234456937274
// MI455X (gfx1250) — compile-verified
//
#include <hip/hip_runtime.h>
#include <hip/hip_bf16.h>

typedef __attribute__((ext_vector_type(2))) float v2f;
typedef __attribute__((ext_vector_type(4))) float v4f;
typedef __attribute__((ext_vector_type(8))) float v8f;

#define BB   8
#define CIN  32
#define HH_  24
#define COUT 64
#define KS   5
#define HO   20
#define LL   400          // HO*HO
#define DD   800          // CIN*KS*KS
#define HP   576          // HH_*HH_

// ---------------- prep: softplus(sp) and W*W flat ----------------
__global__ void vdp_prep_kernel(const float* __restrict__ W,
                                const float* __restrict__ sw,
                                float* __restrict__ wsq,
                                float* __restrict__ sp) {
    int i = blockIdx.x * 256 + threadIdx.x;
    if (i < COUT * CIN * KS * KS) {
        float w = W[i];
        wsq[i] = w * w;
    }
    if (i < COUT) sp[i] = log1pf(expf(sw[i]));
}

// ---------------- im2col of mu and diag(sigma) ----------------
__global__ void vdp_im2col_kernel(const float* __restrict__ mu,
                                  const float* __restrict__ sigma,
                                  float* __restrict__ muM,
                                  float* __restrict__ dsg) {
    int idx = blockIdx.x * 256 + threadIdx.x;
    if (idx >= BB * LL * DD) return;
    int d = idx % DD;
    int l = (idx / DD) % LL;
    int b = idx / (DD * LL);
    int c  = d / (KS * KS);
    int r  = d % (KS * KS);
    int kh = r / KS, kw = r % KS;
    int oh = l / HO, ow = l % HO;
    int h = oh + kh, w = ow + kw;
    muM[idx] = mu[((size_t)(b * CIN + c) * HH_ + h) * HH_ + w];
    int p = h * HH_ + w;
    dsg[idx] = sigma[((size_t)(b * CIN + c) * HP + p) * HP + p];
}

// ---------------- trace[b,l] = sum_d dsg[b,l,d] (one wave per row) ----------------
__global__ void vdp_trace_kernel(const float* __restrict__ dsg,
                                 float* __restrict__ trace) {
    int row  = blockIdx.x * 8 + (threadIdx.x >> 5);   // 8 waves / 256-thread block
    int lane = threadIdx.x & 31;
    if (row >= BB * LL) return;
    const float* p = dsg + (size_t)row * DD;
    float s = 0.f;
    for (int d = lane; d < DD; d += 32) s += p[d];
    for (int off = 16; off > 0; off >>= 1) s += __shfl_xor(s, off, 32);
    if (lane == 0) trace[row] = s;
}

// ---------------- Gram: mmT[b] = mu_mat[b] * mu_mat[b]^T via WMMA f32 16x16x4 ----------------
__global__ void __launch_bounds__(32)
vdp_gram_kernel(const float* __restrict__ muM, float* __restrict__ mmT) {
    int tI = blockIdx.x, tJ = blockIdx.y, b = blockIdx.z;
    int lane = threadIdx.x;
    int half = lane >> 4;          // 0: K={0,1}, 1: K={2,3}
    int r    = lane & 15;
    const float* Ab = muM + ((size_t)b * LL + tI * 16 + r) * DD + half * 2;
    const float* Bb = muM + ((size_t)b * LL + tJ * 16 + r) * DD + half * 2;
    v8f acc = {0.f, 0.f, 0.f, 0.f, 0.f, 0.f, 0.f, 0.f};
    for (int k = 0; k < DD; k += 4) {
        v2f a  = *(const v2f*)(Ab + k);
        v2f bv = *(const v2f*)(Bb + k);
        acc = __builtin_amdgcn_wmma_f32_16x16x4_f32(false, a, false, bv,
                                                    (short)0, acc, false, false);
    }
    // C layout: VGPR v -> M = v + (half?8:0), N = lane&15
    int col     = tJ * 16 + r;
    int rowBase = tI * 16 + (half ? 8 : 0);
    float* o = mmT + ((size_t)b * LL + rowBase) * LL + col;
    #pragma unroll
    for (int v = 0; v < 8; v++) o[(size_t)v * LL] = acc[v];
}

// ---------------- mu_z = mu_mat*W^T  and  mu_w = dsg*w_sq  (16 rows x 64 cols per wave) ----------------
__global__ void __launch_bounds__(32)
vdp_muzw_kernel(const float* __restrict__ muM, const float* __restrict__ dsg,
                const float* __restrict__ Wfull, const float* __restrict__ wsq,
                float* __restrict__ muz, float* __restrict__ muw) {
    int tI = blockIdx.x, b = blockIdx.y;
    int lane = threadIdx.x;
    int half = lane >> 4;
    int r    = lane & 15;
    const float* Amu = muM + ((size_t)b * LL + tI * 16 + r) * DD + half * 2;
    const float* Ads = dsg + ((size_t)b * LL + tI * 16 + r) * DD + half * 2;
    v8f z = {0.f, 0.f, 0.f, 0.f, 0.f, 0.f, 0.f, 0.f};
    v8f accZ[4] = {z, z, z, z};
    v8f accW[4] = {z, z, z, z};
    for (int k = 0; k < DD; k += 4) {
        v2f aMu = *(const v2f*)(Amu + k);
        v2f aDs = *(const v2f*)(Ads + k);
        int kk = k + half * 2;
        #pragma unroll
        for (int nt = 0; nt < 4; nt++) {
            int n = nt * 16 + r;
            // B[k][n] = W[n][k]  -> two contiguous floats of row n
            v2f bW = *(const v2f*)(Wfull + (size_t)n * DD + kk);
            accZ[nt] = __builtin_amdgcn_wmma_f32_16x16x4_f32(false, aMu, false, bW,
                                                             (short)0, accZ[nt], false, false);
            // B[k][n] = wsq_flat[k*COUT + n]   (faithful "buggy" reshape)
            v2f bQ;
            bQ.x = wsq[(size_t)kk * COUT + n];
            bQ.y = wsq[(size_t)(kk + 1) * COUT + n];
            accW[nt] = __builtin_amdgcn_wmma_f32_16x16x4_f32(false, aDs, false, bQ,
                                                             (short)0, accW[nt], false, false);
        }
    }
    #pragma unroll
    for (int nt = 0; nt < 4; nt++) {
        #pragma unroll
        for (int v = 0; v < 8; v++) {
            int lrow = tI * 16 + v + (half ? 8 : 0);
            int c    = nt * 16 + r;
            size_t o = ((size_t)b * COUT + c) * LL + lrow;
            muz[o] = accZ[nt][v];
            muw[o] = accW[nt][v];
        }
    }
}

// ---------------- sigma_z[b,c,l,m] = sp[c]*mmT[b,l,m] (+diag)  -- the 327MB stream ----------------
__global__ void __launch_bounds__(256)
vdp_scale_kernel(const float* __restrict__ mmT, const float* __restrict__ sp,
                 const float* __restrict__ muw, const float* __restrict__ trace,
                 float* __restrict__ out) {
    int bc = blockIdx.y;                 // 0..511
    int b = bc >> 6, c = bc & 63;
    int vec = blockIdx.x * 256 + threadIdx.x;      // float4 index within LxL image
    if (vec >= (LL * LL) / 4) return;
    int l = vec / (LL / 4);
    int m = (vec % (LL / 4)) * 4;
    float s = sp[c];
    v4f mm = *(const v4f*)(mmT + ((size_t)b * LL + l) * LL + m);
    v4f o  = mm * s;
    if (l >= m && l < m + 4) {
        float dterm = muw[((size_t)b * COUT + c) * LL + l] + s * trace[b * LL + l];
        o[l - m] += dterm;
    }
    // non-temporal: keep mmT resident in L2, stream the big output past it
    __builtin_nontemporal_store(o, (v4f*)(out + (size_t)bc * LL * LL + (size_t)l * LL + m));
}

extern "C" void kernel_launch(void* const* d_in, const int* in_sizes, int n_in,
                              void* d_out, int out_size, void* d_ws, size_t ws_size,
                              hipStream_t stream) {
    const float* mu    = (const float*)d_in[0];   // [8,32,24,24]
    const float* sigma = (const float*)d_in[1];   // [8,32,576,576]
    const float* W     = (const float*)d_in[2];   // [64,32,5,5]
    const float* sw    = (const float*)d_in[3];   // [64]
    float* out = (float*)d_out;
    float* ws  = (float*)d_ws;

    float* muM   = ws;                 // 2,560,000
    float* dsg   = ws + 2560000;       // 2,560,000
    float* mmT   = ws + 5120000;       // 1,280,000
    float* muw   = ws + 6400000;       //   204,800
    float* trace = ws + 6604800;       //     3,200
    float* sp    = ws + 6608000;       //        64
    float* wsq   = ws + 6608064;       //    51,200   (total ~26.7 MB)

    float* muz  = out;                 // 204,800 = B*COUT*L
    float* sigz = out + 204800;        // 81,920,000 = B*COUT*L*L

    vdp_prep_kernel<<<(COUT * CIN * KS * KS + 255) / 256, 256, 0, stream>>>(W, sw, wsq, sp);
    vdp_im2col_kernel<<<(BB * LL * DD + 255) / 256, 256, 0, stream>>>(mu, sigma, muM, dsg);
    vdp_trace_kernel<<<(BB * LL) / 8, 256, 0, stream>>>(dsg, trace);
    vdp_gram_kernel<<<dim3(25, 25, 8), 32, 0, stream>>>(muM, mmT);
    vdp_muzw_kernel<<<dim3(25, 8), 32, 0, stream>>>(muM, dsg, W, wsq, muz, muw);
    vdp_scale_kernel<<<dim3((LL * LL / 4 + 255) / 256, BB * COUT), 256, 0, stream>>>(
        mmT, sp, muw, trace, sigz);
}